// CustomLoss_38096359916115
// MI455X (gfx1250) — compile-verified
//
#include <hip/hip_runtime.h>

// ---------------------------------------------------------------------------
// CustomLoss: two bandwidth-bound reductions over 8192x4096 f32 arrays.
//   out[0] = sum( |y|>=mu ? 1 : |y|/mu ) / n_rows
//   out[1] = sum( (y_echo/22.8 - f)^2 ) / n_rows
// Roofline: ~402 MB read @ 23.3 TB/s => ~17us floor; pure streaming problem.
// WMMA (V_WMMA_F32_16X16X4_F32) is used as the per-wave cross-lane reduction
// engine for both partial sums simultaneously (epilogue only, full f32).
// ---------------------------------------------------------------------------

typedef float v2f __attribute__((ext_vector_type(2)));
typedef float v8f __attribute__((ext_vector_type(8)));

#define ECHO_SCALE 22.8f

// Per-wave reduction of a pair (a, b) across 32 lanes.
// Returns: lane 0 -> total(a), lane 1 -> total(b). Other lanes: garbage.
// Requires all 32 lanes active (EXEC all-1s) -- callers guarantee this.
__device__ __forceinline__ float wave_reduce_pair(float a, float b) {
#if defined(__gfx1250__) && __has_builtin(__builtin_amdgcn_wmma_f32_16x16x4_f32)
    const int lane = threadIdx.x & 31;
    // A-matrix (16x4 f32, 2 VGPRs): lane m (and m+16) holds row m.
    //   A[m][0]=a(lane m)   A[m][1]=b(lane m)
    //   A[m][2]=a(lane m+16) A[m][3]=b(lane m+16)
    v2f A; A.x = a; A.y = b;
    // B-matrix (4x16 f32, 2 VGPRs, mirror layout of A):
    //   column 0 selects K=0,2 (the 'a' slots); column 1 selects K=1,3 ('b').
    v2f B;
    B.x = ((lane & 15) == 0) ? 1.0f : 0.0f;   // K=0 (lanes<16) / K=2 (lanes>=16) at N=lane&15
    B.y = ((lane & 15) == 1) ? 1.0f : 0.0f;   // K=1 / K=3 at N=lane&15
    v8f C = {};
    v8f D = __builtin_amdgcn_wmma_f32_16x16x4_f32(
        /*neg_a=*/false, A, /*neg_b=*/false, B,
        /*c_mod=*/(short)0, C, /*reuse_a=*/false, /*reuse_b=*/false);
    // Lane (N=0): VGPR r = D[M=r][0] for lanes 0-15, D[M=8+r][0] for lanes 16-31.
    float s = D[0] + D[1] + D[2] + D[3] + D[4] + D[5] + D[6] + D[7];
    s += __shfl_xor(s, 16, 32);   // fold M=0..7 with M=8..15 halves
    return s;                     // lane0 = sum(a over wave), lane1 = sum(b)
#else
    // Fallback: butterfly shuffle reduction (deterministic, fixed order).
    for (int off = 16; off > 0; off >>= 1) {
        a += __shfl_xor(a, off, 32);
        b += __shfl_xor(b, off, 32);
    }
    return ((threadIdx.x & 31) == 1) ? b : a;
#endif
}

__global__ void __launch_bounds__(256)
loss_partials_kernel(const float* __restrict__ y,
                     const float* __restrict__ y_echo,
                     const float* __restrict__ f,
                     const float* __restrict__ mu_p,
                     float* __restrict__ ws,
                     long long total) {
    const float mu       = mu_p[0];
    const float inv_mu   = 1.0f / mu;
    const float inv_echo = 1.0f / ECHO_SCALE;

    const long long total4 = total >> 2;
    const long long stride = (long long)gridDim.x * blockDim.x;
    const long long tid    = (long long)blockIdx.x * blockDim.x + threadIdx.x;

    const float4* __restrict__ y4 = (const float4*)y;
    const float4* __restrict__ e4 = (const float4*)y_echo;
    const float4* __restrict__ f4 = (const float4*)f;

    float l0 = 0.0f, l2 = 0.0f;

    // Streaming phase: 3 independent b128 loads per iteration, grid-stride.
    for (long long i = tid; i < total4; i += stride) {
        float4 yv = y4[i];
        float4 ev = e4[i];
        float4 fv = f4[i];

        float ay, d;
        ay = fabsf(yv.x); l0 += (ay >= mu) ? 1.0f : ay * inv_mu;
        ay = fabsf(yv.y); l0 += (ay >= mu) ? 1.0f : ay * inv_mu;
        ay = fabsf(yv.z); l0 += (ay >= mu) ? 1.0f : ay * inv_mu;
        ay = fabsf(yv.w); l0 += (ay >= mu) ? 1.0f : ay * inv_mu;

        d = fmaf(ev.x, inv_echo, -fv.x); l2 = fmaf(d, d, l2);
        d = fmaf(ev.y, inv_echo, -fv.y); l2 = fmaf(d, d, l2);
        d = fmaf(ev.z, inv_echo, -fv.z); l2 = fmaf(d, d, l2);
        d = fmaf(ev.w, inv_echo, -fv.w); l2 = fmaf(d, d, l2);
    }

    // Scalar tail (total % 4), handled once.
    if (tid == 0) {
        for (long long i = (total4 << 2); i < total; ++i) {
            float ay = fabsf(y[i]);
            l0 += (ay >= mu) ? 1.0f : ay * inv_mu;
            float d = fmaf(y_echo[i], inv_echo, -f[i]);
            l2 = fmaf(d, d, l2);
        }
    }

    // Epilogue: WMMA per-wave reduction of both sums at once.
    float s = wave_reduce_pair(l0, l2);

    __shared__ float lds[16];                 // 8 waves x {l0, l2}
    const int wid  = threadIdx.x >> 5;
    const int lane = threadIdx.x & 31;
    if (lane < 2) lds[wid * 2 + lane] = s;
    __syncthreads();

    if (threadIdx.x < 2) {                    // thread 0 -> l0, thread 1 -> l2
        float acc = 0.0f;
        const int nwaves = blockDim.x >> 5;
        for (int w = 0; w < nwaves; ++w) acc += lds[w * 2 + threadIdx.x];
        ws[blockIdx.x * 2 + threadIdx.x] = acc;
    }
}

__global__ void __launch_bounds__(256)
loss_finalize_kernel(const float* __restrict__ ws,
                     float* __restrict__ out,
                     int nblocks, float scale) {
    float l0 = 0.0f, l2 = 0.0f;
    for (int i = threadIdx.x; i < nblocks; i += blockDim.x) {
        l0 += ws[2 * i];
        l2 += ws[2 * i + 1];
    }

    float s = wave_reduce_pair(l0, l2);

    __shared__ float lds[16];
    const int wid  = threadIdx.x >> 5;
    const int lane = threadIdx.x & 31;
    if (lane < 2) lds[wid * 2 + lane] = s;
    __syncthreads();

    if (threadIdx.x < 2) {
        float acc = 0.0f;
        const int nwaves = blockDim.x >> 5;
        for (int w = 0; w < nwaves; ++w) acc += lds[w * 2 + threadIdx.x];
        out[threadIdx.x] = acc * scale;       // out[0]=pseudo_l0, out[1]=l2
    }
}

extern "C" void kernel_launch(void* const* d_in, const int* in_sizes, int n_in,
                              void* d_out, int out_size, void* d_ws, size_t ws_size,
                              hipStream_t stream) {
    const float* y      = (const float*)d_in[0];
    const float* y_echo = (const float*)d_in[1];
    const float* f      = (const float*)d_in[2];
    const float* mu     = (const float*)d_in[3];
    float*       out    = (float*)d_out;
    float*       ws     = (float*)d_ws;

    const long long total  = (long long)in_sizes[0];   // 8192*4096
    const long long n_rows = total / 4096;             // reference divides by n=8192
    const float     scale  = 1.0f / (float)n_rows;

    const int NB = 2048;   // 2048 block-partial pairs (16 KB in d_ws)
    const int NT = 256;    // 8 wave32s per block

    loss_partials_kernel<<<NB, NT, 0, stream>>>(y, y_echo, f, mu, ws, total);
    loss_finalize_kernel<<<1, NT, 0, stream>>>(ws, out, NB, scale);
}